// SPVBlock_41360535061139
// MI455X (gfx1250) — compile-verified
//
#include <hip/hip_runtime.h>

#define NV   120000
#define ND   30000
#define NV2  60000
#define NPTS 400000
#define CC   128
#define HH   64

typedef float v2f __attribute__((ext_vector_type(2)));
typedef float v8f __attribute__((ext_vector_type(8)));

__device__ __forceinline__ float lrelu(float v) { return v > 0.f ? v : 0.1f * v; }

__device__ __forceinline__ v8f wmma4(v2f a, v2f b, v8f c) {
    // D = A(16x4) * B(4x16) + C, fp32 exact
    return __builtin_amdgcn_wmma_f32_16x16x4_f32(false, a, false, b, (short)0, c, false, false);
}

// Async copy 16B from global to LDS (per-lane), tracked by ASYNCcnt.
__device__ __forceinline__ void async_copy16(unsigned lds_off, const float* g) {
    asm volatile("global_load_async_to_lds_b128 %0, %1, off"
                 :: "v"(lds_off), "v"(g) : "memory");
}
__device__ __forceinline__ void wait_async0() {
    asm volatile("s_wait_asynccnt 0" ::: "memory");
}
__device__ __forceinline__ unsigned lds_off32(const void* p) {
    return (unsigned)(unsigned long long)p;   // low 32 bits of generic = LDS byte offset
}

// A row pointer and Bt column pointer are pre-offset by 2*(lane>>4)
__device__ __forceinline__ v8f wloop(const float* __restrict__ Ar,
                                     const float* __restrict__ Bc, int K, v8f acc) {
    for (int kk = 0; kk < K; kk += 4) {
        v2f a = *(const v2f*)(Ar + kk);
        v2f b = *(const v2f*)(Bc + kk);
        acc = wmma4(a, b, acc);
    }
    return acc;
}

// Same but applies per-k affine (BatchNorm scale/shift) to A while loading.
__device__ __forceinline__ v8f wloop_bn(const float* __restrict__ Ar,
                                        const float* __restrict__ sc,
                                        const float* __restrict__ sh,
                                        const float* __restrict__ Bc, int K, v8f acc) {
    for (int kk = 0; kk < K; kk += 4) {
        v2f a = *(const v2f*)(Ar + kk);
        v2f s = *(const v2f*)(sc + kk);
        v2f t = *(const v2f*)(sh + kk);
        a.x = a.x * s.x + t.x;
        a.y = a.y * s.y + t.y;
        v2f b = *(const v2f*)(Bc + kk);
        acc = wmma4(a, b, acc);
    }
    return acc;
}

// ---------- elementwise / scatter kernels ----------

__global__ __launch_bounds__(256) void add_scatter_kernel(
        const float* __restrict__ features, const float* __restrict__ vfea,
        const int* __restrict__ inv, float* __restrict__ x,
        float* __restrict__ ds, float* __restrict__ ds_cnt) {
    int tid = blockIdx.x * 256 + threadIdx.x;      // over NV*32 float4s
    int v = tid >> 5, q = tid & 31;
    float4 f = ((const float4*)features)[tid];
    float4 g = ((const float4*)vfea)[tid];
    f.x += g.x; f.y += g.y; f.z += g.z; f.w += g.w;
    ((float4*)x)[tid] = f;
    int d = inv[v];
    float* dst = ds + (size_t)d * CC + q * 4;
    atomicAdd(dst + 0, f.x); atomicAdd(dst + 1, f.y);
    atomicAdd(dst + 2, f.z); atomicAdd(dst + 3, f.w);
    if (q == 0) atomicAdd(ds_cnt + d, 1.0f);
}

__global__ __launch_bounds__(256) void div_rows_kernel(
        float* __restrict__ buf, const float* __restrict__ cnt) {
    int tid = blockIdx.x * 256 + threadIdx.x;      // over nrows*32 float4s
    float c = 1.0f / fmaxf(cnt[tid >> 5], 1.0f);
    float4 v = ((float4*)buf)[tid];
    v.x *= c; v.y *= c; v.z *= c; v.w *= c;
    ((float4*)buf)[tid] = v;
}

__global__ __launch_bounds__(256) void transpose_kernel(
        const float* __restrict__ src, float* __restrict__ dst, int K, int N) {
    int tid = blockIdx.x * 256 + threadIdx.x;
    if (tid >= K * N) return;
    int k = tid / N, n = tid % N;
    dst[n * K + k] = src[tid];
}

__global__ void bn_fin_kernel(const float* __restrict__ sum, const float* __restrict__ sq,
                              const float* __restrict__ g, const float* __restrict__ beta,
                              float* __restrict__ scale, float* __restrict__ shift) {
    int t = threadIdx.x;   // 64 threads
    float m = sum[t] * (1.0f / (float)ND);
    float v = sq[t] * (1.0f / (float)ND) - m * m;
    float s = g[t] * rsqrtf(v + 1e-5f);
    scale[t] = s;
    shift[t] = beta[t] - m * s;
}

// ---------- WMMA GEMM kernels ----------

// N=64 GEMM (ND rows), fused lrelu+bias + BN stat accumulation. 4 waves/block.
__global__ __launch_bounds__(128) void gemm_h_kernel(
        const float* __restrict__ A, int K,
        const float* __restrict__ sc, const float* __restrict__ sh,
        const float* __restrict__ Bt, const float* __restrict__ bias,
        float* __restrict__ out, float* __restrict__ gsum, float* __restrict__ gsq) {
    int lane = threadIdx.x & 31, w = threadIdx.x >> 5;
    int hi = lane >> 4, l16 = lane & 15;
    int row = blockIdx.x * 16 + l16;
    int col = w * 16 + l16;
    const float* Ar = A + (size_t)row * K + 2 * hi;
    const float* Bc = Bt + (size_t)col * K + 2 * hi;
    v8f acc = {0.f, 0.f, 0.f, 0.f, 0.f, 0.f, 0.f, 0.f};
    if (sc) acc = wloop_bn(Ar, sc + 2 * hi, sh + 2 * hi, Bc, K, acc);
    else    acc = wloop(Ar, Bc, K, acc);

    __shared__ float s_sum[64], s_sq[64];
    if (threadIdx.x < 64) { s_sum[threadIdx.x] = 0.f; s_sq[threadIdx.x] = 0.f; }
    __syncthreads();

    float b = bias[col];
    float ls = 0.f, lq = 0.f;
    #pragma unroll
    for (int j = 0; j < 8; ++j) {
        int m = j + 8 * hi;
        float val = lrelu(acc[j] + b);
        out[(size_t)(blockIdx.x * 16 + m) * HH + col] = val;
        ls += val; lq += val * val;
    }
    atomicAdd(&s_sum[col], ls);
    atomicAdd(&s_sq[col], lq);
    __syncthreads();
    if (threadIdx.x < 64) {
        atomicAdd(&gsum[threadIdx.x], s_sum[threadIdx.x]);
        atomicAdd(&gsq[threadIdx.x], s_sq[threadIdx.x]);
    }
}

// N=128 GEMM, 16 rows/block (used for the ND-row layer 3; ND % 32 != 0).
__global__ __launch_bounds__(256) void gemm_wide_kernel(
        const float* __restrict__ A, int K,
        const float* __restrict__ sc, const float* __restrict__ sh,
        const float* __restrict__ Bt, const float* __restrict__ bias,
        float* __restrict__ out) {
    int lane = threadIdx.x & 31, w = threadIdx.x >> 5;
    int hi = lane >> 4, l16 = lane & 15;
    int row = blockIdx.x * 16 + l16;
    int col = w * 16 + l16;
    const float* Ar = A + (size_t)row * K + 2 * hi;
    const float* Bc = Bt + (size_t)col * K + 2 * hi;
    v8f acc = {0.f, 0.f, 0.f, 0.f, 0.f, 0.f, 0.f, 0.f};
    if (sc) acc = wloop_bn(Ar, sc + 2 * hi, sh + 2 * hi, Bc, K, acc);
    else    acc = wloop(Ar, Bc, K, acc);
    float b = bias[col];
    #pragma unroll
    for (int j = 0; j < 8; ++j) {
        int m = j + 8 * hi;
        out[(size_t)(blockIdx.x * 16 + m) * CC + col] = lrelu(acc[j] + b);
    }
}

// identity = lrelu(x @ W_in + b_in): 32 rows x 128 cols per block, A staged in
// LDS via async copies, 2 M-tiles per wave sharing each B fragment.
#define IDS 132   // LDS row stride (132 % 64 == 4 -> conflict-free; 132*4 % 16 == 0)
__global__ __launch_bounds__(256) void gemm_id_kernel(
        const float* __restrict__ A, const float* __restrict__ Bt,
        const float* __restrict__ bias, float* __restrict__ out) {
    __shared__ float s_a[32 * IDS];
    int t = threadIdx.x;
    int base = blockIdx.x * 32;
    // stage A: 8 threads/row, 16 floats (4x16B) each
    {
        int r = t >> 3, c0 = (t & 7) * 16;
        const float* src = A + (size_t)(base + r) * CC + c0;
        unsigned dst = lds_off32(&s_a[r * IDS + c0]);
        #pragma unroll
        for (int i = 0; i < 4; ++i) async_copy16(dst + 16 * i, src + 4 * i);
    }
    wait_async0();
    __syncthreads();

    int lane = t & 31, w = t >> 5;
    int hi = lane >> 4, l16 = lane & 15;
    int col = w * 16 + l16;
    const float* Bc = Bt + (size_t)col * CC + 2 * hi;
    const float* Ar = &s_a[l16 * IDS + 2 * hi];
    v8f acc0 = {0.f, 0.f, 0.f, 0.f, 0.f, 0.f, 0.f, 0.f};
    v8f acc1 = acc0;
    for (int kk = 0; kk < CC; kk += 4) {
        v2f b = *(const v2f*)(Bc + kk);
        v2f a0 = *(const v2f*)(Ar + kk);
        v2f a1 = *(const v2f*)(Ar + 16 * IDS + kk);
        acc0 = wmma4(a0, b, acc0);
        acc1 = wmma4(a1, b, acc1);
    }
    float b = bias[col];
    #pragma unroll
    for (int j = 0; j < 8; ++j) {
        int m = j + 8 * hi;
        out[(size_t)(base + m) * CC + col]      = lrelu(acc0[j] + b);
        out[(size_t)(base + 16 + m) * CC + col] = lrelu(acc1[j] + b);
    }
}

// Fused point MLP: 64 points x 128 cols per block. Gathered K=256 A-tile
// async-copied to LDS; 4 M-tiles per wave share each B fragment; hidden tile
// re-staged in the same LDS; second GEMM; atomic scatter into p_sum/p_cnt.
#define PAS 260   // A stride   (260 % 64 == 4; 260*4 % 16 == 0)
#define PHS 132   // hidden stride
__global__ __launch_bounds__(256) void gemm_point_kernel(
        const float* __restrict__ identity, const float* __restrict__ hm,
        const int* __restrict__ inv, const int* __restrict__ cil,
        const int* __restrict__ cin,
        const float* __restrict__ Wo1t, const float* __restrict__ bo1,
        const float* __restrict__ Wo2t, const float* __restrict__ bo2,
        float* __restrict__ p_sum, float* __restrict__ p_cnt) {
    __shared__ float s_buf[64 * PAS];
    __shared__ int s_seg[64];
    int t = threadIdx.x;
    int base = blockIdx.x * 64;
    if (t < 64) s_seg[t] = cin[base + t];

    // stage gathered A (64 rows x 256): 4 threads/row, 64 floats (16x16B) each.
    // chunk c0 in {0,64,128,192} lies entirely in identity- or hm-half.
    {
        int r = t >> 2, c0 = (t & 3) * 64;
        int v = cil[base + r];
        const float* src = (c0 < CC) ? identity + (size_t)v * CC + c0
                                     : hm + (size_t)inv[v] * CC + (c0 - CC);
        unsigned dst = lds_off32(&s_buf[r * PAS + c0]);
        #pragma unroll
        for (int i = 0; i < 16; ++i) async_copy16(dst + 16 * i, src + 4 * i);
    }
    wait_async0();
    __syncthreads();

    int lane = t & 31, w = t >> 5;
    int hi = lane >> 4, l16 = lane & 15;
    int col = w * 16 + l16;

    // GEMM1: hidden = lrelu(A @ Wo1 + bo1), K = 256
    const float* B1 = Wo1t + (size_t)col * 256 + 2 * hi;
    const float* Ar = &s_buf[l16 * PAS + 2 * hi];
    v8f acc[4];
    #pragma unroll
    for (int mt = 0; mt < 4; ++mt) acc[mt] = {0.f, 0.f, 0.f, 0.f, 0.f, 0.f, 0.f, 0.f};
    for (int kk = 0; kk < 256; kk += 4) {
        v2f b = *(const v2f*)(B1 + kk);
        #pragma unroll
        for (int mt = 0; mt < 4; ++mt) {
            v2f a = *(const v2f*)(Ar + mt * (16 * PAS) + kk);
            acc[mt] = wmma4(a, b, acc[mt]);
        }
    }
    float b1 = bo1[col];
    __syncthreads();                       // everyone done reading A tile
    #pragma unroll
    for (int mt = 0; mt < 4; ++mt)
        #pragma unroll
        for (int j = 0; j < 8; ++j) {
            int m = mt * 16 + j + 8 * hi;
            s_buf[m * PHS + col] = lrelu(acc[mt][j] + b1);
        }
    __syncthreads();

    // GEMM2: out = hidden @ Wo2 + bo2, K = 128
    const float* B2 = Wo2t + (size_t)col * CC + 2 * hi;
    const float* Hr = &s_buf[l16 * PHS + 2 * hi];
    v8f acc2[4];
    #pragma unroll
    for (int mt = 0; mt < 4; ++mt) acc2[mt] = {0.f, 0.f, 0.f, 0.f, 0.f, 0.f, 0.f, 0.f};
    for (int kk = 0; kk < CC; kk += 4) {
        v2f b = *(const v2f*)(B2 + kk);
        #pragma unroll
        for (int mt = 0; mt < 4; ++mt) {
            v2f a = *(const v2f*)(Hr + mt * (16 * PHS) + kk);
            acc2[mt] = wmma4(a, b, acc2[mt]);
        }
    }
    float b2 = bo2[col];
    #pragma unroll
    for (int mt = 0; mt < 4; ++mt)
        #pragma unroll
        for (int j = 0; j < 8; ++j) {
            int m = mt * 16 + j + 8 * hi;
            atomicAdd(&p_sum[(size_t)s_seg[m] * CC + col], acc2[mt][j] + b2);
        }
    if (t < 64) atomicAdd(&p_cnt[s_seg[t]], 1.0f);
}

__global__ __launch_bounds__(256) void gather_out_kernel(
        const float* __restrict__ p_fea, const int* __restrict__ cin,
        float* __restrict__ out) {
    int tid = blockIdx.x * 256 + threadIdx.x;   // over NPTS*32 float4s
    int p = tid >> 5, q = tid & 31;
    int s = cin[p];
    ((float4*)out)[tid] = ((const float4*)p_fea)[(size_t)s * 32 + q];
}

// ---------- host launcher ----------

#define F_DS     ((size_t)0)                    // ND*CC       (zeroed)
#define F_DSCNT  (F_DS + (size_t)ND * CC)       // ND          (zeroed)
#define F_PSUM   (F_DSCNT + ND)                 // NV2*CC      (zeroed)
#define F_PCNT   (F_PSUM + (size_t)NV2 * CC)    // NV2         (zeroed)
#define F_SUM1   (F_PCNT + NV2)                 // 4x64        (zeroed)
#define F_SQ1    (F_SUM1 + 64)
#define F_SUM2   (F_SQ1 + 64)
#define F_SQ2    (F_SUM2 + 64)
#define F_ZEND   (F_SQ2 + 64)
#define F_X      (F_ZEND)
#define F_H1     (F_X + (size_t)NV * CC)
#define F_H2     (F_H1 + (size_t)ND * HH)
#define F_HM     (F_H2 + (size_t)ND * HH)
#define F_ID     (F_HM + (size_t)ND * CC)
#define F_WINT   (F_ID + (size_t)NV * CC)
#define F_W1T    (F_WINT + CC * CC)
#define F_W2T    (F_W1T + CC * HH)
#define F_W3T    (F_W2T + HH * HH)
#define F_WO1T   (F_W3T + HH * CC)
#define F_WO2T   (F_WO1T + 2 * CC * CC)
#define F_SC1    (F_WO2T + CC * CC)
#define F_SH1    (F_SC1 + 64)
#define F_SC2    (F_SH1 + 64)
#define F_SH2    (F_SC2 + 64)

extern "C" void kernel_launch(void* const* d_in, const int* in_sizes, int n_in,
                              void* d_out, int out_size, void* d_ws, size_t ws_size,
                              hipStream_t stream) {
    (void)in_sizes; (void)n_in; (void)out_size; (void)ws_size;
    const float* features = (const float*)d_in[0];
    const float* vfea     = (const float*)d_in[1];
    const int*   inv      = (const int*)d_in[2];
    const int*   cil      = (const int*)d_in[3];
    const int*   cin      = (const int*)d_in[4];
    const float* W_in = (const float*)d_in[5];  const float* b_in = (const float*)d_in[6];
    const float* W1 = (const float*)d_in[7];    const float* b1 = (const float*)d_in[8];
    const float* g1 = (const float*)d_in[9];    const float* be1 = (const float*)d_in[10];
    const float* W2 = (const float*)d_in[11];   const float* b2 = (const float*)d_in[12];
    const float* g2 = (const float*)d_in[13];   const float* be2 = (const float*)d_in[14];
    const float* W3 = (const float*)d_in[15];   const float* b3 = (const float*)d_in[16];
    const float* Wo1 = (const float*)d_in[17];  const float* bo1 = (const float*)d_in[18];
    const float* Wo2 = (const float*)d_in[19];  const float* bo2 = (const float*)d_in[20];
    float* out = (float*)d_out;
    float* ws  = (float*)d_ws;

    hipMemsetAsync(ws, 0, F_ZEND * sizeof(float), stream);

    transpose_kernel<<<(CC * CC + 255) / 256, 256, 0, stream>>>(W_in, ws + F_WINT, CC, CC);
    transpose_kernel<<<(CC * HH + 255) / 256, 256, 0, stream>>>(W1, ws + F_W1T, CC, HH);
    transpose_kernel<<<(HH * HH + 255) / 256, 256, 0, stream>>>(W2, ws + F_W2T, HH, HH);
    transpose_kernel<<<(HH * CC + 255) / 256, 256, 0, stream>>>(W3, ws + F_W3T, HH, CC);
    transpose_kernel<<<(2 * CC * CC + 255) / 256, 256, 0, stream>>>(Wo1, ws + F_WO1T, 2 * CC, CC);
    transpose_kernel<<<(CC * CC + 255) / 256, 256, 0, stream>>>(Wo2, ws + F_WO2T, CC, CC);

    add_scatter_kernel<<<NV * 32 / 256, 256, 0, stream>>>(
        features, vfea, inv, ws + F_X, ws + F_DS, ws + F_DSCNT);
    div_rows_kernel<<<ND * 32 / 256, 256, 0, stream>>>(ws + F_DS, ws + F_DSCNT);

    gemm_h_kernel<<<ND / 16, 128, 0, stream>>>(
        ws + F_DS, CC, nullptr, nullptr, ws + F_W1T, b1, ws + F_H1, ws + F_SUM1, ws + F_SQ1);
    bn_fin_kernel<<<1, 64, 0, stream>>>(ws + F_SUM1, ws + F_SQ1, g1, be1, ws + F_SC1, ws + F_SH1);
    gemm_h_kernel<<<ND / 16, 128, 0, stream>>>(
        ws + F_H1, HH, ws + F_SC1, ws + F_SH1, ws + F_W2T, b2, ws + F_H2, ws + F_SUM2, ws + F_SQ2);
    bn_fin_kernel<<<1, 64, 0, stream>>>(ws + F_SUM2, ws + F_SQ2, g2, be2, ws + F_SC2, ws + F_SH2);
    gemm_wide_kernel<<<ND / 16, 256, 0, stream>>>(
        ws + F_H2, HH, ws + F_SC2, ws + F_SH2, ws + F_W3T, b3, ws + F_HM);

    gemm_id_kernel<<<NV / 32, 256, 0, stream>>>(ws + F_X, ws + F_WINT, b_in, ws + F_ID);

    gemm_point_kernel<<<NPTS / 64, 256, 0, stream>>>(
        ws + F_ID, ws + F_HM, inv, cil, cin,
        ws + F_WO1T, bo1, ws + F_WO2T, bo2, ws + F_PSUM, ws + F_PCNT);

    div_rows_kernel<<<NV2 * 32 / 256, 256, 0, stream>>>(ws + F_PSUM, ws + F_PCNT);
    gather_out_kernel<<<NPTS * 32 / 256, 256, 0, stream>>>(ws + F_PSUM, cin, out);
}